// CubemapLight_16149077033155
// MI455X (gfx1250) — compile-verified
//
#include <hip/hip_runtime.h>
#include <stdint.h>

// ---------------------------------------------------------------------------
// Cubemap mip-chain build, fused single-pass version for MI455X (gfx1250).
// Base [6,2048,2048,3] f32 -> levels 2048,1024,...,16 concatenated in d_out.
// Pure bandwidth problem: read base once via async-to-LDS DMA, produce all
// 7 pooled levels from LDS, forward mip0 via async-from-LDS stores.
// ---------------------------------------------------------------------------

#define FACES      6
#define RES0       2048
#define TILE       128                   // base pixels per tile edge
#define TPF        (RES0 / TILE)         // 16 tiles per face edge
#define NCHUNK     8                     // 8 chunks of 16 rows = 128 rows
#define CHROWS     16
#define ROWF       (TILE * 3)            // 384 floats per tile row
#define FACE_ROWF  (RES0 * 3)            // 6144 floats per face row
#define FACE_ELEMS ((size_t)RES0 * RES0 * 3)

// The async-to-LDS builtins take pointers to 16-byte int vectors
// (per the round-1 diagnostic: "__vector_size__(4*sizeof(int)) int *").
typedef int v4i __attribute__((ext_vector_type(4)));
typedef __attribute__((address_space(1))) v4i gv4i;   // global-memory side
typedef __attribute__((address_space(3))) v4i lv4i;   // LDS side

// ---- CDNA5 async Global<->LDS path (probe via __has_builtin) --------------
#if defined(__has_builtin)
#if __has_builtin(__builtin_amdgcn_global_load_async_to_lds_b128) && \
    __has_builtin(__builtin_amdgcn_global_store_async_from_lds_b128)
#define HAVE_ASYNC 1
#endif
#endif
#ifndef HAVE_ASYNC
#define HAVE_ASYNC 0
#if defined(__AMDGCN__)
#pragma message("gfx1250 device pass: async-to-LDS builtins NOT found; synchronous fallback in use")
#endif
#endif

__device__ __forceinline__ gv4i* as_global_v4(const void* p) {
  return (gv4i*)(uintptr_t)p;
}
__device__ __forceinline__ lv4i* as_lds_v4(const void* p) {
  // generic LDS address = {aperture, offset32}; low 32 bits are the AS(3) offset
  return (lv4i*)(uint32_t)(uintptr_t)p;
}

__device__ __forceinline__ void async_load16(const float* g, float* l) {
#if HAVE_ASYNC
  __builtin_amdgcn_global_load_async_to_lds_b128(as_global_v4(g), as_lds_v4(l), 0, 0);
#else
  *(float4*)l = *(const float4*)g;
#endif
}

__device__ __forceinline__ void async_store16(float* g, const float* l) {
#if HAVE_ASYNC
  __builtin_amdgcn_global_store_async_from_lds_b128(as_global_v4(g), as_lds_v4(l), 0, 0);
#else
  *(float4*)g = *(const float4*)l;
#endif
}

template <int N>
__device__ __forceinline__ void wait_async() {
#if HAVE_ASYNC
#if __has_builtin(__builtin_amdgcn_s_wait_asynccnt)
  __builtin_amdgcn_s_wait_asynccnt(N);
#else
  asm volatile("s_wait_asynccnt %0" ::"i"(N) : "memory");
#endif
#endif
}

// ---------------------------------------------------------------------------

__global__ __launch_bounds__(256) void mip_pyramid_kernel(
    const float* __restrict__ base, float* __restrict__ dout) {
  __shared__ __align__(16) float stage[2][CHROWS * ROWF];  // 2 x 24 KB staging
  __shared__ __align__(16) float bufB[64 * 192];           // 48 KB (level-1 tile)
  __shared__ __align__(16) float bufC[32 * 96];            // 12 KB (ping-pong)

  const int tid   = threadIdx.x;
  const int b     = blockIdx.x;
  const int face  = b >> 8;      // / (TPF*TPF)
  const int trem  = b & 255;
  const int tileY = trem >> 4;
  const int tileX = trem & 15;

  // flat float offsets of each mip level in d_out
  const size_t offs[8] = {0ull,         75497472ull,  94371840ull,  99090432ull,
                          100270080ull, 100564992ull, 100638720ull, 100657152ull};

  const float* fbase = base + (size_t)face * FACE_ELEMS +
                       (size_t)(tileY * TILE) * FACE_ROWF + (size_t)tileX * ROWF;
  float* f0out = dout + (size_t)face * FACE_ELEMS +
                 (size_t)(tileY * TILE) * FACE_ROWF + (size_t)tileX * ROWF;

  // ---- prologue: issue async loads for chunk 0 (1536 x 16B, 6 per thread) --
#pragma unroll
  for (int j = 0; j < 6; ++j) {
    int i = tid + j * 256;          // 0..1535
    int row = i / 96;               // 0..15
    int seg = i - row * 96;         // 16-byte segment within the 1536B tile row
    async_load16(fbase + (size_t)row * FACE_ROWF + seg * 4,
                 &stage[0][row * ROWF + seg * 4]);
  }

  for (int k = 0; k < NCHUNK; ++k) {
    // Each wave drains its OWN async ops (loads of chunk k, stores of chunk
    // k-1) before the barrier; the barrier then makes all waves' LDS traffic
    // visible and the stale buffer reusable.
    wait_async<0>();
    __syncthreads();

    // ---- issue async loads for chunk k+1 (overlaps with compute below) ----
    if (k + 1 < NCHUNK) {
      const float* gsrc = fbase + (size_t)(k + 1) * CHROWS * FACE_ROWF;
      float* lbuf = stage[(k + 1) & 1];
#pragma unroll
      for (int j = 0; j < 6; ++j) {
        int i = tid + j * 256;
        int row = i / 96;
        int seg = i - row * 96;
        async_load16(gsrc + (size_t)row * FACE_ROWF + seg * 4,
                     lbuf + row * ROWF + seg * 4);
      }
    }

    // ---- level-1 pooling for chunk k: 16 rows -> 8 rows x 64 px x 3 ch ----
    {
      const float* s = stage[k & 1];
      float* g1 = dout + offs[1] + (size_t)face * (1024ull * 1024 * 3) +
                  (size_t)(tileY * 64 + k * 8) * 3072 + (size_t)tileX * 192;
#pragma unroll
      for (int j = 0; j < 6; ++j) {
        int i = tid + j * 256;       // 0..1535 output floats in this chunk
        int lr  = i / 192;           // level-1 row within chunk (0..7)
        int idx = i - lr * 192;      // float within row (64 px * 3 ch)
        int c   = idx / 3;
        int chn = idx - c * 3;
        int x0  = (2 * lr) * ROWF + 6 * c + chn;
        float v = 0.25f * (s[x0] + s[x0 + 3] + s[x0 + ROWF] + s[x0 + ROWF + 3]);
        bufB[(k * 8 + lr) * 192 + idx] = v;
        g1[(size_t)lr * 3072 + idx]    = v;
      }
    }

    // ---- forward chunk k of the base to mip0 via async LDS->global stores -
    {
      float* gdst = f0out + (size_t)k * CHROWS * FACE_ROWF;
      const float* lsrc = stage[k & 1];
#pragma unroll
      for (int j = 0; j < 6; ++j) {
        int i = tid + j * 256;
        int row = i / 96;
        int seg = i - row * 96;
        async_store16(gdst + (size_t)row * FACE_ROWF + seg * 4,
                      lsrc + row * ROWF + seg * 4);
      }
    }
    __syncthreads();
  }

  // ---- levels 2..7 entirely in LDS (tile shrinks 64 -> 1 px) --------------
  const float* pin = bufB;
  float* pout = bufC;
#pragma unroll
  for (int L = 2; L <= 7; ++L) {
    const int Win  = 128 >> (L - 1);
    const int Wout = Win >> 1;
    const int n    = Wout * Wout * 3;
    const int res  = RES0 >> L;
    const int tl   = TILE >> L;
    const int si   = Win * 3;
    float* g = dout + offs[L] + (size_t)face * ((size_t)res * res * 3) +
               (size_t)(tileY * tl) * (res * 3) + (size_t)(tileX * tl) * 3;
    for (int i = tid; i < n; i += 256) {
      int r   = i / (Wout * 3);
      int rem = i - r * (Wout * 3);
      int c   = rem / 3;
      int chn = rem - c * 3;
      int x0  = 2 * r * si + 6 * c + chn;
      float v = 0.25f * (pin[x0] + pin[x0 + 3] + pin[x0 + si] + pin[x0 + si + 3]);
      pout[i] = v;
      g[(size_t)r * (res * 3) + rem] = v;
    }
    __syncthreads();
    const float* t = pout;
    pout = (float*)pin;
    pin  = t;
  }
}

extern "C" void kernel_launch(void* const* d_in, const int* in_sizes, int n_in,
                              void* d_out, int out_size, void* d_ws, size_t ws_size,
                              hipStream_t stream) {
  (void)in_sizes; (void)n_in; (void)out_size; (void)d_ws; (void)ws_size;
  const float* base = (const float*)d_in[0];
  float* out = (float*)d_out;
  // 6 faces x 16 x 16 tiles of 128x128 base pixels
  mip_pyramid_kernel<<<dim3(FACES * TPF * TPF), dim3(256), 0, stream>>>(base, out);
}